// Encoder_28509992911015
// MI455X (gfx1250) — compile-verified
//
#include <hip/hip_runtime.h>

typedef __attribute__((ext_vector_type(2))) float v2f;
typedef __attribute__((ext_vector_type(8))) float v8f;

#define DIM   128
#define LDSP  132   // padded LDS row stride (floats): 132 % 64 == 4 -> conflict-free column reads

// ---------------------------------------------------------------------------
// Zero the aggregation buffer (can't memset safely under graph capture rules).
// ---------------------------------------------------------------------------
__global__ void gin_zero(float4* __restrict__ p, int n4) {
    int i = blockIdx.x * blockDim.x + threadIdx.x;
    if (i < n4) p[i] = make_float4(0.f, 0.f, 0.f, 0.f);
}

// ---------------------------------------------------------------------------
// Neighbor-sum scatter: one wave32 per edge. 32 lanes x float4 = one 128-float
// row. Coalesced 512B gather of z[src], 4 consecutive f32 atomics per lane
// into agg[dst] (resolved in L2 -- agg is 25.6MB, L2 is 192MB).
// ---------------------------------------------------------------------------
__global__ void gin_scatter(const float* __restrict__ z,
                            const int*  __restrict__ src,
                            const int*  __restrict__ dst,
                            float* __restrict__ agg, int nEdges) {
    int wave = (blockIdx.x * blockDim.x + threadIdx.x) >> 5;
    int lane = threadIdx.x & 31;
    if (wave >= nEdges) return;
    int s = src[wave];
    int d = dst[wave];
    float4 v = ((const float4*)(z + (size_t)s * DIM))[lane];
    float* ap = agg + (size_t)d * DIM + lane * 4;
    atomicAdd(ap + 0, v.x);
    atomicAdd(ap + 1, v.y);
    atomicAdd(ap + 2, v.z);
    atomicAdd(ap + 3, v.w);
}

// ---------------------------------------------------------------------------
// Fused GIN MLP for one 16-row node tile:
//   zout = relu( relu((agg+zin) @ W1 + b1) @ W2 + b2 )
// 256 threads = 8 waves; wave w owns output columns [16w, 16w+16) of both
// GEMMs via V_WMMA_F32_16X16X4_F32 (exact fp32, K-loop of 32 per GEMM).
// A fragment layout (32-bit 16x4): lane 0-15 -> M=lane, {K, K+1};
//                                  lane 16-31 -> M=lane-16, {K+2, K+3}.
// C/D layout (v8f): row = r + 8*(lane>=16), col = lane&15.
// ---------------------------------------------------------------------------
__global__ __launch_bounds__(256) void gin_mlp(
    const float* __restrict__ agg, const float* __restrict__ zin,
    const float* __restrict__ W1,  const float* __restrict__ b1,
    const float* __restrict__ W2,  const float* __restrict__ b2,
    float* __restrict__ zout, int nNodes)
{
    __shared__ float As[16 * LDSP];
    __shared__ float Xs[16 * LDSP];

    const int rowBase = blockIdx.x * 16;
    const int tid = threadIdx.x;

    // Stage A = agg + zin (16 x 128) into padded LDS; 8 floats per thread.
    for (int i = tid; i < 16 * DIM; i += 256) {
        int r = i >> 7;           // i / 128
        int cidx = i & (DIM - 1);
        int gr = rowBase + r;
        float v = 0.f;
        if (gr < nNodes)
            v = agg[(size_t)gr * DIM + cidx] + zin[(size_t)gr * DIM + cidx];
        As[r * LDSP + cidx] = v;
    }
    __syncthreads();

    const int wave = tid >> 5;
    const int lane = tid & 31;
    const int mn   = lane & 15;          // M for A-frag, N for B-frag / C cols
    const int kh   = (lane >> 4) << 1;   // 0 or 2: K sub-offset per half-wave
    const int colBase = wave * 16;
    const int mrow = (lane >> 4) << 3;   // C/D row base: 0 or 8

    // ---- GEMM1: (agg+z) @ W1 ----
    v8f c1 = {};
    const float* W1c = W1 + colBase;
    #pragma unroll 4
    for (int k = 0; k < DIM; k += 4) {
        v2f a, b;
        a.x = As[mn * LDSP + k + kh];
        a.y = As[mn * LDSP + k + kh + 1];
        b.x = W1c[(size_t)(k + kh) * DIM + mn];
        b.y = W1c[(size_t)(k + kh + 1) * DIM + mn];
        c1 = __builtin_amdgcn_wmma_f32_16x16x4_f32(
                 false, a, false, b, (short)0, c1, false, false);
    }
    // bias + relu -> Xs
    {
        float bias = b1[colBase + mn];
        #pragma unroll
        for (int r = 0; r < 8; ++r) {
            float v = c1[r] + bias;
            Xs[(mrow + r) * LDSP + colBase + mn] = v > 0.f ? v : 0.f;
        }
    }
    __syncthreads();

    // ---- GEMM2: X1 @ W2 ----
    v8f c2 = {};
    const float* W2c = W2 + colBase;
    #pragma unroll 4
    for (int k = 0; k < DIM; k += 4) {
        v2f a, b;
        a.x = Xs[mn * LDSP + k + kh];
        a.y = Xs[mn * LDSP + k + kh + 1];
        b.x = W2c[(size_t)(k + kh) * DIM + mn];
        b.y = W2c[(size_t)(k + kh + 1) * DIM + mn];
        c2 = __builtin_amdgcn_wmma_f32_16x16x4_f32(
                 false, a, false, b, (short)0, c2, false, false);
    }
    // bias + relu -> global zout
    {
        float bias = b2[colBase + mn];
        #pragma unroll
        for (int r = 0; r < 8; ++r) {
            int gr = rowBase + mrow + r;
            if (gr < nNodes) {
                float v = c2[r] + bias;
                zout[(size_t)gr * DIM + colBase + mn] = v > 0.f ? v : 0.f;
            }
        }
    }
}

// ---------------------------------------------------------------------------
// Driver: 3 layers of { zero agg; scatter; fused MLP }.
// Inputs: x, Ws1[3,128,128], bs1[3,128], Ws2[3,128,128], bs2[3,128],
//         edge_index[2,E] (int32).
// ---------------------------------------------------------------------------
extern "C" void kernel_launch(void* const* d_in, const int* in_sizes, int n_in,
                              void* d_out, int out_size, void* d_ws, size_t ws_size,
                              hipStream_t stream) {
    const float* x   = (const float*)d_in[0];
    const float* Ws1 = (const float*)d_in[1];
    const float* bs1 = (const float*)d_in[2];
    const float* Ws2 = (const float*)d_in[3];
    const float* bs2 = (const float*)d_in[4];
    const int*   ei  = (const int*)d_in[5];

    const int nNodes = in_sizes[0] / DIM;
    const int nEdges = in_sizes[5] / 2;
    const int nLayers = in_sizes[1] / (DIM * DIM);

    const int* src = ei;            // edge_index[0]
    const int* dst = ei + nEdges;   // edge_index[1]

    const size_t nd = (size_t)nNodes * DIM;
    float* agg = (float*)d_ws;
    float* zA  = agg + nd;
    float* zB  = zA + nd;

    const int n4 = (int)(nd / 4);
    const int zeroBlocks    = (n4 + 255) / 256;
    const int scatterBlocks = (nEdges * 32 + 255) / 256;   // one wave32 per edge
    const int mlpBlocks     = (nNodes + 15) / 16;          // one 16-row tile per block

    const float* zin = x;
    for (int l = 0; l < nLayers; ++l) {
        float* zout = (l == nLayers - 1) ? (float*)d_out : ((l & 1) ? zB : zA);

        gin_zero<<<zeroBlocks, 256, 0, stream>>>((float4*)agg, n4);
        gin_scatter<<<scatterBlocks, 256, 0, stream>>>(zin, src, dst, agg, nEdges);
        gin_mlp<<<mlpBlocks, 256, 0, stream>>>(
            agg, zin,
            Ws1 + (size_t)l * DIM * DIM, bs1 + (size_t)l * DIM,
            Ws2 + (size_t)l * DIM * DIM, bs2 + (size_t)l * DIM,
            zout, nNodes);

        zin = zout;
    }
}